// ObjectQuerySelector_39599598469221
// MI455X (gfx1250) — compile-verified
//
#include <hip/hip_runtime.h>
#include <hip/hip_bf16.h>

// Problem constants (from the reference)
constexpr int B = 8;
constexpr int N = 20000;
constexpr int D = 256;
constexpr int C = 91;
constexpr int K = 300;

typedef unsigned uint32x4 __attribute__((ext_vector_type(4)));
typedef int      int32x4  __attribute__((ext_vector_type(4)));
typedef int      int32x8  __attribute__((ext_vector_type(8)));

// Address-space-qualified pointer types for the gfx1250 async LDS builtins.
// (Round-1 diagnostic showed param0 is `int32x4 __device__ *`, i.e. AS1;
//  the LDS-side param is the __shared__/AS3 counterpart.)
typedef __attribute__((address_space(1))) int32x4* as1_v4i;
typedef __attribute__((address_space(3))) int32x4* as3_v4i;

#if __has_builtin(__builtin_amdgcn_global_load_async_to_lds_b128) && \
    __has_builtin(__builtin_amdgcn_global_store_async_from_lds_b128) && \
    __has_builtin(__builtin_amdgcn_s_wait_asynccnt)
#define USE_ASYNC_LDS_COPY 1
#else
#define USE_ASYNC_LDS_COPY 0
#endif

// ---------------------------------------------------------------------------
// Kernel 1: per-row max over C=91 logits -> monotonic sortable u32 key.
// One wave32 per row; 8 rows per 256-thread block. Coalesced streaming reads.
// ---------------------------------------------------------------------------
__global__ __launch_bounds__(256)
void score_kernel(const float* __restrict__ logits, unsigned* __restrict__ keys) {
    const int wave = threadIdx.x >> 5;
    const int lane = threadIdx.x & 31;
    const long long row = (long long)blockIdx.x * 8 + wave;   // [0, B*N)
    if (row >= (long long)B * N) return;

    const float* p = logits + row * C;
    // Prefetch the rows the *next* block will read (lowers to global_prefetch_b8).
    __builtin_prefetch(p + 8LL * C);

    float m = -__builtin_inff();
#pragma unroll
    for (int c = lane; c < C; c += 32) m = fmaxf(m, p[c]);

    // wave32 max-reduce
#pragma unroll
    for (int off = 16; off > 0; off >>= 1)
        m = fmaxf(m, __shfl_xor(m, off, 32));

    if (lane == 0) {
        unsigned bits = __float_as_uint(m);
        // order-preserving float -> uint transform (handles negatives)
        unsigned u = bits ^ (((unsigned)((int)bits >> 31)) | 0x80000000u);
        keys[row] = u;
    }
}

// ---------------------------------------------------------------------------
// Kernel 2: exact top-K per batch (JAX top_k semantics: score desc, ties ->
// lowest index first). One 1024-thread workgroup per batch.
//   Phase A: 4-pass LDS-histogram radix select -> u* (K-th largest key) and
//            `need` = #elements equal to u* that belong in the top-K.
//   Phase B: collect keys > u*  (count = K-need <= 299), bitonic-sort the
//            (key, N-1-idx) 64-bit pairs descending -> exact order.
//   Phase C: stable block-wide compaction in index order of keys == u*,
//            keep the first `need`.
// ---------------------------------------------------------------------------
__global__ __launch_bounds__(1024)
void select_kernel(const unsigned* __restrict__ keys_all, int* __restrict__ sel_all) {
    __shared__ unsigned hist[256];
    __shared__ unsigned long long cand[512];
    __shared__ int scan[1024];
    __shared__ unsigned s_prefix, s_need, s_cntA;
    __shared__ int s_base;

    const int b = blockIdx.x;
    const unsigned* keys = keys_all + (size_t)b * N;
    int* sel = sel_all + b * K;
    const int tid = threadIdx.x;

    if (tid == 0) { s_prefix = 0u; s_need = (unsigned)K; s_cntA = 0u; }
    __syncthreads();

    // ---- Phase A: radix select (MSD, 8 bits per pass) ----
    for (int shift = 24; shift >= 0; shift -= 8) {
        if (tid < 256) hist[tid] = 0u;
        __syncthreads();
        const unsigned prefix = s_prefix;
        const unsigned maskHi = (shift == 24) ? 0u : (0xFFFFFFFFu << (shift + 8));
        for (int n = tid; n < N; n += 1024) {
            unsigned u = keys[n];
            if ((u & maskHi) == prefix)
                atomicAdd(&hist[(u >> shift) & 255u], 1u);
        }
        __syncthreads();
        if (tid == 0) {
            unsigned need = s_need, cum = 0u;
            int d = 255;
            for (; d >= 0; --d) {
                unsigned h = hist[d];
                if (cum + h >= need) { s_need = need - cum; break; }
                cum += h;
            }
            s_prefix = prefix | ((unsigned)d << shift);
        }
        __syncthreads();
    }
    const unsigned ustar = s_prefix;

    // ---- Phase B: strictly-greater set, sorted exactly ----
    for (int i = tid; i < 512; i += 1024) cand[i] = 0ull;
    __syncthreads();
    for (int n = tid; n < N; n += 1024) {
        unsigned u = keys[n];
        if (u > ustar) {
            unsigned p = atomicAdd(&s_cntA, 1u);
            // key desc primary, index asc secondary (N-1-n so bigger == smaller idx)
            cand[p] = ((unsigned long long)u << 32) | (unsigned)(N - 1 - n);
        }
    }
    __syncthreads();
    const int cntA = (int)s_cntA;          // == K - s_need, <= 299

    // bitonic sort cand[0..511] descending (pad zeros sink to the end)
    for (int kk = 2; kk <= 512; kk <<= 1) {
        for (int j = kk >> 1; j > 0; j >>= 1) {
            const int i = tid;
            if (i < 512) {
                const int ixj = i ^ j;
                if (ixj > i) {
                    unsigned long long a = cand[i], c = cand[ixj];
                    const bool doSwap = ((i & kk) == 0) ? (a < c) : (a > c);
                    if (doSwap) { cand[i] = c; cand[ixj] = a; }
                }
            }
            __syncthreads();
        }
    }
    for (int r = tid; r < cntA; r += 1024)
        sel[r] = (N - 1) - (int)(unsigned)(cand[r] & 0xFFFFFFFFu);

    // ---- Phase C: equal-to-u* set, lowest indices first ----
    const int needB = (int)s_need;         // >= 1
    if (tid == 0) s_base = 0;
    __syncthreads();
    for (int start = 0; start < N; start += 1024) {
        const int base = s_base;           // uniform after barrier
        if (base >= needB) break;
        const int n = start + tid;
        const int flag = (n < N && keys[n] == ustar) ? 1 : 0;
        scan[tid] = flag;
        __syncthreads();
        // Hillis-Steele inclusive scan over 1024
        for (int off = 1; off < 1024; off <<= 1) {
            int v = (tid >= off) ? scan[tid - off] : 0;
            __syncthreads();
            scan[tid] += v;
            __syncthreads();
        }
        const int pos = base + scan[tid] - flag;   // exclusive rank
        if (flag && pos < needB) sel[cntA + pos] = n;
        __syncthreads();
        if (tid == 0) s_base = base + scan[1023];
        __syncthreads();
    }
}

// ---------------------------------------------------------------------------
// Kernel 3: gather. One 128-thread block per (batch, rank).
// The 256-float memory row uses the CDNA5 async global<->LDS copy path
// (global_load_async_to_lds_b128 -> s_wait_asynccnt -> store_async_from_lds):
// zero VGPR traffic for the bulk copy. Logits (91f) + coords (4f) are plain.
// ---------------------------------------------------------------------------
__global__ __launch_bounds__(128)
void gather_kernel(const float* __restrict__ memory,
                   const float* __restrict__ logits,
                   const float* __restrict__ coords,
                   const int* __restrict__ sel,
                   float* __restrict__ out) {
    const int br = blockIdx.x;             // [0, B*K)
    const int b = br / K, r = br % K;
    const int idx = sel[br];
    const int tid = threadIdx.x;

    float* out_mem = out;                                   // B*K*D
    float* out_log = out + (size_t)B * K * D;               // B*K*C
    float* out_crd = out_log + (size_t)B * K * C;           // B*K*4

    const float4* src4 = (const float4*)(memory + ((size_t)b * N + idx) * D);
    float4*       dst4 = (float4*)(out_mem + ((size_t)b * K + r) * D);

#if USE_ASYNC_LDS_COPY
    __shared__ int32x4 rowbuf[D / 4];      // 64 x 16B = one memory row
    if (tid < D / 4) {
        // each lane copies 16 bytes from global src4 + tid into LDS rowbuf[tid] (ASYNCcnt)
        __builtin_amdgcn_global_load_async_to_lds_b128(
            (as1_v4i)(src4 + tid), (as3_v4i)&rowbuf[tid], 0, 0);
        // Async-load LDS writes and async-store LDS reads are NOT ordered
        // with each other (ISA 10.x) -> drain ASYNCcnt first.
        __builtin_amdgcn_s_wait_asynccnt(0);
        // each lane copies 16 bytes from LDS rowbuf[tid] to global dst4 + tid (ASYNCcnt)
        __builtin_amdgcn_global_store_async_from_lds_b128(
            (as1_v4i)(dst4 + tid), (as3_v4i)&rowbuf[tid], 0, 0);
        // S_ENDPGM performs an implicit wait-idle, draining the store.
    }
#else
    if (tid < D / 4) dst4[tid] = src4[tid];
#endif

    const float* srcl = logits + ((size_t)b * N + idx) * C;
    float*       dstl = out_log + ((size_t)b * K + r) * C;
    if (tid < C) dstl[tid] = srcl[tid];

    const float* srcc = coords + ((size_t)b * N + idx) * 4;
    float*       dstc = out_crd + ((size_t)b * K + r) * 4;
    if (tid < 4) dstc[tid] = srcc[tid];
}

// ---------------------------------------------------------------------------
// Probe kernel (NEVER launched): confirms the TDM / cluster paths lower on
// this toolchain (6-arg tensor_load_to_lds per round-1 diagnostic).
// ---------------------------------------------------------------------------
__global__ void cdna5_probe_kernel(const float* g, float* o) {
    __shared__ float lds[64];
#if __has_builtin(__builtin_amdgcn_tensor_load_to_lds)
    {
        uint32x4 g0 = {};
        int32x8  g1 = {};
        int32x4  g2 = {}, g3 = {};
        int32x8  g4 = {};
        __builtin_amdgcn_tensor_load_to_lds(g0, g1, g2, g3, g4, 0);
    }
#endif
#if __has_builtin(__builtin_amdgcn_s_wait_tensorcnt)
    __builtin_amdgcn_s_wait_tensorcnt(0);
#endif
#if __has_builtin(__builtin_amdgcn_s_cluster_barrier)
    __builtin_amdgcn_s_cluster_barrier();
#endif
    __syncthreads();
    if (o) o[0] = lds[0] + (g ? g[0] : 0.f);
}

// ---------------------------------------------------------------------------
extern "C" void kernel_launch(void* const* d_in, const int* in_sizes, int n_in,
                              void* d_out, int out_size, void* d_ws, size_t ws_size,
                              hipStream_t stream) {
    (void)in_sizes; (void)n_in; (void)out_size; (void)ws_size;

    const float* memory = (const float*)d_in[0];   // (B, N, D) f32
    const float* logits = (const float*)d_in[1];   // (B, N, C) f32
    const float* coords = (const float*)d_in[2];   // (B, N, 4) f32
    float* out = (float*)d_out;                    // mem | logits | coords, flat

    unsigned* keys = (unsigned*)d_ws;                                  // B*N u32
    int* sel = (int*)((char*)d_ws + (size_t)B * N * sizeof(unsigned)); // B*K i32

    score_kernel<<<(B * N + 7) / 8, 256, 0, stream>>>(logits, keys);
    select_kernel<<<B, 1024, 0, stream>>>(keys, sel);
    gather_kernel<<<B * K, 128, 0, stream>>>(memory, logits, coords, sel, out);
}